// GraphSAGEConv_30700426232191
// MI455X (gfx1250) — compile-verified
//
#include <hip/hip_runtime.h>

// GraphSAGE fused pipeline for MI455X (gfx1250, wave32).
//
// Algebraic restructuring:
//   h_neigh = mean_s(node_feat[src]) + mean_s(edge_feat) @ W_edge + b_edge
//   out     = relu( node_feat@W1 + nbar@W2 + ebar@(W_edge@W2) + (b_edge@W2 + b_lin) )
// This turns the [1.6M x 64] edge GEMM into a pure streaming reduction (410 MB,
// non-temporal so L2 keeps node_feat resident for the gather) plus a small
// [N x 320] x [320 x 128] fp32 WMMA GEMM. Bandwidth-bound: ~0.6 GB @ 23.3 TB/s.

#define N_NODES  100000
#define SAMP     16
#define NODE_DIM 128
#define EDGE_DIM 64
#define EMB_DIM  128

typedef float v2f __attribute__((ext_vector_type(2)));
typedef float v8f __attribute__((ext_vector_type(8)));

// ---------------------------------------------------------------------------
// Kernel A: per-node mean of gathered src-node features and of edge features.
// One 128-thread block per node. src_idx loads are block-uniform -> scalar
// loads; node_feat gather rows are 512B coalesced and L2-resident (51 MB);
// edge_feat is streamed once with non-temporal hint (410 MB).
// ---------------------------------------------------------------------------
__global__ __launch_bounds__(128) void sage_reduce(
    const float* __restrict__ node_feat,   // [N,128]
    const float* __restrict__ edge_feat,   // [E,64]
    const int*   __restrict__ src_idx,     // [E]
    float* __restrict__ nbar,              // [N,128]
    float* __restrict__ ebar)              // [N,64]
{
    const int  n    = blockIdx.x;
    const int  t    = threadIdx.x;
    const long base = (long)n * SAMP;

    float nsum = 0.f;
    float esum = 0.f;
#pragma unroll
    for (int s = 0; s < SAMP; ++s) {
        const int idx = src_idx[base + s];                       // uniform -> s_load
        nsum += node_feat[(long)idx * NODE_DIM + t];
        if (t < EDGE_DIM)                                        // waves 0-1 fully active
            esum += __builtin_nontemporal_load(
                        &edge_feat[(base + s) * EDGE_DIM + t]);  // streaming, NT
    }
    nbar[(long)n * NODE_DIM + t] = nsum * (1.f / SAMP);
    if (t < EDGE_DIM)
        ebar[(long)n * EDGE_DIM + t] = esum * (1.f / SAMP);
}

// ---------------------------------------------------------------------------
// Kernel B (tiny): Wcomb = W_edge @ W_lin[128:256,:]   (64x128)
//                  bcomb = b_edge @ W_lin[128:256,:] + b_lin
// ---------------------------------------------------------------------------
__global__ __launch_bounds__(128) void sage_prep_w(
    const float* __restrict__ W_edge,   // [64,128]
    const float* __restrict__ b_edge,   // [128]
    const float* __restrict__ W_lin,    // [256,128]
    const float* __restrict__ b_lin,    // [128]
    float* __restrict__ Wcomb,          // [64,128]
    float* __restrict__ bcomb)          // [128]
{
    const int j = threadIdx.x;
    const float* W2 = W_lin + NODE_DIM * EMB_DIM;   // bottom half rows

    float bc = b_lin[j];
    for (int c = 0; c < NODE_DIM; ++c)
        bc += b_edge[c] * W2[c * EMB_DIM + j];
    bcomb[j] = bc;

    for (int k = 0; k < EDGE_DIM; ++k) {
        float acc = 0.f;
        for (int c = 0; c < NODE_DIM; ++c)
            acc += W_edge[k * NODE_DIM + c] * W2[c * EMB_DIM + j];  // scalar x coalesced
        Wcomb[k * EMB_DIM + j] = acc;
    }
}

// ---------------------------------------------------------------------------
// Kernel C: fused fp32 WMMA GEMM + bias + ReLU.
//   out[n, j] = relu( node_feat[n]@W1 + nbar[n]@W2 + ebar[n]@Wcomb + bcomb )
// 256 threads = 8 wave32; block covers 16 nodes x 128 cols; each wave owns one
// 16x16 tile (col = wave*16) and accumulates K = 128+128+64 via
// v_wmma_f32_16x16x4_f32 (80 WMMAs/wave). No divergence -> EXEC all ones.
//
// fp32 16x4 A layout (ISA 7.12.2): lanes 0-15 hold M=0..15; VGPR0/1 = K{0,1}
// for lanes 0-15 and K{2,3} for lanes 16-31. B (4x16) mirrors with N across
// lanes. C/D: VGPR r -> row M = r + 8*laneHalf, col N = lane%16.
// ---------------------------------------------------------------------------
__global__ __launch_bounds__(256) void sage_gemm(
    const float* __restrict__ node_feat, // [N,128]
    const float* __restrict__ nbar,      // [N,128]
    const float* __restrict__ ebar,      // [N,64]
    const float* __restrict__ W_lin,     // [256,128] (top half = W1)
    const float* __restrict__ Wcomb,     // [64,128]
    const float* __restrict__ bcomb,     // [128]
    float* __restrict__ out)             // [N,128]
{
    const int tile = blockIdx.x * 16;          // base node of this block
    const int wave = threadIdx.x >> 5;         // 0..7
    const int lane = threadIdx.x & 31;
    const int hl   = lane >> 4;                // lane half: 0 or 1
    const int m    = lane & 15;                // M (for A) / N (for B) index
    const int col  = wave * 16;                // output column base

    v8f c = {};

    // ---- phase 1: node_feat @ W1, K = 128 ----
    {
        const float* X = node_feat + (long)(tile + m) * NODE_DIM;
        const float* W = W_lin;
#pragma unroll 8
        for (int k0 = 0; k0 < NODE_DIM; k0 += 4) {
            const int ka = k0 + 2 * hl;
            v2f a; a.x = X[ka];                    a.y = X[ka + 1];          // contiguous pair
            v2f b; b.x = W[ka * EMB_DIM + col + m]; b.y = W[(ka + 1) * EMB_DIM + col + m];
            c = __builtin_amdgcn_wmma_f32_16x16x4_f32(false, a, false, b, (short)0, c, false, false);
        }
    }
    // ---- phase 2: nbar @ W2, K = 128 ----
    {
        const float* X = nbar + (long)(tile + m) * NODE_DIM;
        const float* W = W_lin + NODE_DIM * EMB_DIM;
#pragma unroll 8
        for (int k0 = 0; k0 < NODE_DIM; k0 += 4) {
            const int ka = k0 + 2 * hl;
            v2f a; a.x = X[ka];                    a.y = X[ka + 1];
            v2f b; b.x = W[ka * EMB_DIM + col + m]; b.y = W[(ka + 1) * EMB_DIM + col + m];
            c = __builtin_amdgcn_wmma_f32_16x16x4_f32(false, a, false, b, (short)0, c, false, false);
        }
    }
    // ---- phase 3: ebar @ Wcomb, K = 64 ----
    {
        const float* X = ebar + (long)(tile + m) * EDGE_DIM;
        const float* W = Wcomb;
#pragma unroll 8
        for (int k0 = 0; k0 < EDGE_DIM; k0 += 4) {
            const int ka = k0 + 2 * hl;
            v2f a; a.x = X[ka];                    a.y = X[ka + 1];
            v2f b; b.x = W[ka * EMB_DIM + col + m]; b.y = W[(ka + 1) * EMB_DIM + col + m];
            c = __builtin_amdgcn_wmma_f32_16x16x4_f32(false, a, false, b, (short)0, c, false, false);
        }
    }

    // ---- epilogue: bias + ReLU, scatter 8 rows ----
    const float bias = bcomb[col + m];
#pragma unroll
    for (int r = 0; r < 8; ++r) {
        const int row = tile + r + 8 * hl;
        float v = c[r] + bias;
        out[(long)row * EMB_DIM + col + m] = v > 0.f ? v : 0.f;
    }
}

// ---------------------------------------------------------------------------
extern "C" void kernel_launch(void* const* d_in, const int* in_sizes, int n_in,
                              void* d_out, int out_size, void* d_ws, size_t ws_size,
                              hipStream_t stream) {
    const float* node_feat = (const float*)d_in[0];
    const float* edge_feat = (const float*)d_in[1];
    const int*   src_idx   = (const int*)  d_in[2];
    const float* W_edge    = (const float*)d_in[3];
    const float* b_edge    = (const float*)d_in[4];
    const float* W_lin     = (const float*)d_in[5];
    const float* b_lin     = (const float*)d_in[6];

    // workspace layout: nbar [N,128] | ebar [N,64] | Wcomb [64,128] | bcomb [128]
    char*  ws    = (char*)d_ws;
    float* nbar  = (float*)ws;
    float* ebar  = (float*)(ws + (size_t)N_NODES * NODE_DIM * sizeof(float));
    float* Wcomb = (float*)(ws + (size_t)N_NODES * (NODE_DIM + EDGE_DIM) * sizeof(float));
    float* bcomb = Wcomb + EDGE_DIM * EMB_DIM;

    sage_reduce<<<N_NODES, 128, 0, stream>>>(node_feat, edge_feat, src_idx, nbar, ebar);
    sage_prep_w<<<1, 128, 0, stream>>>(W_edge, b_edge, W_lin, b_lin, Wcomb, bcomb);
    sage_gemm<<<N_NODES / 16, 256, 0, stream>>>(node_feat, nbar, ebar, W_lin, Wcomb, bcomb,
                                                (float*)d_out);
}